// Transformer_82935818486079
// MI455X (gfx1250) — compile-verified
//
#include <hip/hip_runtime.h>
#include <cstddef>

// ---------------- problem constants ----------------
#define KDIM   76
#define KP     80      // K padded to 5x16
#define HEADS  8
#define TSEQ   1024
#define BATCH  8
#define DEPTH  6
#define HK     608     // HEADS*KDIM
#define F4     304
#define F4P    320     // 4K padded to 20x16
#define NCLASS 2

typedef float v2f __attribute__((ext_vector_type(2)));
typedef float v8f __attribute__((ext_vector_type(8)));
typedef int   v4i __attribute__((vector_size(16)));   // matches builtin param type

// ---------------- WMMA f32 16x16x4 wrapper ----------------
__device__ __forceinline__ v8f wmma4(v2f a, v2f b, v8f c) {
#if __has_builtin(__builtin_amdgcn_wmma_f32_16x16x4_f32)
  return __builtin_amdgcn_wmma_f32_16x16x4_f32(false, a, false, b, (short)0, c,
                                               false, false);
#else
  c[0] += a.x * b.x + a.y * b.y;  // host-pass placeholder
  return c;
#endif
}

__device__ __forceinline__ void store_tile(float* out, int ldo, v8f c) {
  const int lane = threadIdx.x & 31;
  const int n  = lane & 15;
  const int mh = (lane >> 4) << 3;   // 0 or 8
#pragma unroll
  for (int j = 0; j < 8; ++j) out[(mh + j) * ldo + n] = c[j];
}

// ---------------- async global->LDS staging (gfx1250 path) ----------------
#if defined(__gfx1250__) && __has_builtin(__builtin_amdgcn_global_load_async_to_lds_b128)
#define HAVE_ASYNC_LDS 1
#else
#define HAVE_ASYNC_LDS 0
#endif

__device__ __forceinline__ void cp16_g2l(const float* g, float* l) {
#if HAVE_ASYNC_LDS
  __attribute__((address_space(1))) v4i* gp =
      (__attribute__((address_space(1))) v4i*)const_cast<float*>(g);
  __attribute__((address_space(3))) v4i* lp =
      (__attribute__((address_space(3))) v4i*)l;
  __builtin_amdgcn_global_load_async_to_lds_b128(gp, lp, 0, 0);
#else
  *(float4*)l = *(const float4*)g;
#endif
}

__device__ __forceinline__ void wait_async_all() {
#if HAVE_ASYNC_LDS
#if __has_builtin(__builtin_amdgcn_s_wait_asynccnt)
  __builtin_amdgcn_s_wait_asynccnt(0);
#else
  asm volatile("s_wait_asynccnt 0" ::: "memory");
#endif
#endif
}

// ============================================================
// Weight packing into WMMA B-chunk order:
//   Bp[((nt*(Kc/4) + c)*32 + lane)*2 + e]
//     = W[4c + 2*(lane>>4) + e][nt*16 + (lane&15)]   (0 outside valid range)
// so each lane's two k-values per chunk are one contiguous b64.
// ============================================================

// Wq/Wk/Wv: (D, K, HK) -> packed (D, H, 5, 20, 32, 2)
__global__ void pack_qkv_w(const float* __restrict__ W, float* __restrict__ Wp) {
  int i = blockIdx.x * blockDim.x + threadIdx.x;
  const int total = DEPTH * HEADS * KP * KP;
  if (i >= total) return;
  int e = i & 1; int r = i >> 1;
  int lane = r & 31; r >>= 5;
  int c  = r % 20; r /= 20;
  int nt = r % 5;  r /= 5;
  int h  = r % HEADS; int d = r / HEADS;
  int kp = c * 4 + ((lane >> 4) << 1) + e;
  int np = nt * 16 + (lane & 15);
  float v = 0.f;
  if (kp < KDIM && np < KDIM) v = W[(d * KDIM + kp) * HK + h * KDIM + np];
  Wp[i] = v;
}

// Wu: (D, HK, K) -> packed (D, H, 5, 20, 32, 2)
__global__ void pack_wu(const float* __restrict__ W, float* __restrict__ Wp) {
  int i = blockIdx.x * blockDim.x + threadIdx.x;
  const int total = DEPTH * HEADS * KP * KP;
  if (i >= total) return;
  int e = i & 1; int r = i >> 1;
  int lane = r & 31; r >>= 5;
  int c  = r % 20; r /= 20;
  int nt = r % 5;  r /= 5;
  int h  = r % HEADS; int d = r / HEADS;
  int kp = c * 4 + ((lane >> 4) << 1) + e;
  int np = nt * 16 + (lane & 15);
  float v = 0.f;
  if (kp < KDIM && np < KDIM) v = W[(d * HK + h * KDIM + kp) * KDIM + np];
  Wp[i] = v;
}

// W1: (D, K, F4) -> packed (D, 20, 20, 32, 2)   (Kc=KP, Ntiles=20)
__global__ void pack_w1(const float* __restrict__ W, float* __restrict__ Wp) {
  int i = blockIdx.x * blockDim.x + threadIdx.x;
  const int total = DEPTH * KP * F4P;
  if (i >= total) return;
  int e = i & 1; int r = i >> 1;
  int lane = r & 31; r >>= 5;
  int c  = r % 20; r /= 20;
  int nt = r % 20; r /= 20;
  int d  = r;
  int kp = c * 4 + ((lane >> 4) << 1) + e;
  int np = nt * 16 + (lane & 15);
  float v = 0.f;
  if (kp < KDIM && np < F4) v = W[(d * KDIM + kp) * F4 + np];
  Wp[i] = v;
}

// W2: (D, F4, K) -> packed (D, 5, 80, 32, 2)    (Kc=F4P, Ntiles=5)
__global__ void pack_w2(const float* __restrict__ W, float* __restrict__ Wp) {
  int i = blockIdx.x * blockDim.x + threadIdx.x;
  const int total = DEPTH * F4P * KP;
  if (i >= total) return;
  int e = i & 1; int r = i >> 1;
  int lane = r & 31; r >>= 5;
  int c  = r % 80; r /= 80;
  int nt = r % 5;  r /= 5;
  int d  = r;
  int kp = c * 4 + ((lane >> 4) << 1) + e;
  int np = nt * 16 + (lane & 15);
  float v = 0.f;
  if (kp < F4 && np < KDIM) v = W[(d * F4 + kp) * KDIM + np];
  Wp[i] = v;
}

// ---------------- embed: concat of 4 linears + pos ----------------
__global__ void embed_kernel(const float* __restrict__ v, const float* __restrict__ x,
                             const float* __restrict__ y, const float* __restrict__ z,
                             const float* __restrict__ Whp, const float* __restrict__ bhp,
                             const float* __restrict__ Wlm, const float* __restrict__ blm,
                             const float* __restrict__ Wel, const float* __restrict__ bel,
                             const float* __restrict__ Wau, const float* __restrict__ bau,
                             const float* __restrict__ pos, float* __restrict__ h) {
  const int token = blockIdx.x;             // 0..B*T-1
  const int t = token % TSEQ;
  const int col = threadIdx.x;
  if (col >= KP) return;
  if (col >= KDIM) { h[(size_t)token * KP + col] = 0.f; return; }
  float acc;
  if (col < 6) {
    acc = bhp[col];
    const float* r = v + (size_t)token * 6;
    for (int j = 0; j < 6; ++j) acc += r[j] * Whp[j * 6 + col];
  } else if (col < 36) {
    int c2 = col - 6; acc = blm[c2];
    const float* r = x + (size_t)token * 137;
    for (int j = 0; j < 137; ++j) acc += r[j] * Wlm[j * 30 + c2];
  } else if (col < 66) {
    int c2 = col - 36; acc = bel[c2];
    const float* r = y + (size_t)token * 113;
    for (int j = 0; j < 113; ++j) acc += r[j] * Wel[j * 30 + c2];
  } else {
    int c2 = col - 66; acc = bau[c2];
    const float* r = z + (size_t)token * 35;
    for (int j = 0; j < 35; ++j) acc += r[j] * Wau[j * 10 + c2];
  }
  h[(size_t)token * KP + col] = acc + pos[t * KDIM + col];
}

// ---------------- Q/K/V projection: wave computes a full 16x80 head panel ----
// grid (T/16, HEADS, B), block 32.  A chunk loaded once, reused by 5 WMMAs.
__global__ __launch_bounds__(32) void qkv_kernel(const float* __restrict__ hbuf,
                                                 const float* __restrict__ Wp,
                                                 float* __restrict__ out) {
  const int tt = blockIdx.x * 16;
  const int h  = blockIdx.y;
  const int b  = blockIdx.z;
  const int lane = threadIdx.x & 31;
  const int m  = lane & 15;
  const int kb = (lane >> 4) << 1;
  const float* A  = hbuf + ((size_t)b * TSEQ + tt) * KP;
  const float* Bp = Wp + (size_t)h * KP * KP;
  v8f acc[5];
#pragma unroll
  for (int nt = 0; nt < 5; ++nt) acc[nt] = v8f{};
  for (int c = 0; c < 20; ++c) {
    v2f a;
    a.x = A[m * KP + c * 4 + kb];
    a.y = A[m * KP + c * 4 + kb + 1];
#pragma unroll
    for (int nt = 0; nt < 5; ++nt) {
      v2f bv = *(const v2f*)(Bp + ((size_t)(nt * 20 + c) * 32 + lane) * 2);
      acc[nt] = wmma4(a, bv, acc[nt]);
    }
  }
  float* O = out + (((size_t)(b * HEADS + h)) * TSEQ + tt) * KP;
#pragma unroll
  for (int nt = 0; nt < 5; ++nt) store_tile(O + nt * 16, KP, acc[nt]);
}

// ---------------- fused flash attention ----------------
// grid (T/128, H, B), block 256 = 8 waves; each wave owns one 16-query tile.
// K/V tiles double-buffered in LDS via async global->LDS copies.
__global__ __launch_bounds__(256) void attn_kernel(const float* __restrict__ Qb,
                                                   const float* __restrict__ Kb,
                                                   const float* __restrict__ Vb,
                                                   float* __restrict__ Ob) {
  __shared__ float ldsK[2][16 * KP];
  __shared__ float ldsV[2][16 * KP];
  __shared__ float ldsP[8][16 * 17];
  const int lane = threadIdx.x & 31;
  const int wave = threadIdx.x >> 5;
  const int m  = lane & 15;
  const int kb = (lane >> 4) << 1;
  const int mh = (lane >> 4) << 3;
  const int b = blockIdx.z, h = blockIdx.y;
  const int t0 = (blockIdx.x * 8 + wave) * 16;
  const size_t base = ((size_t)(b * HEADS + h)) * TSEQ * KP;
  const float scale = 0.11470786693528087f;  // 1/sqrt(76)

  // Q tile (16xKP) into A-layout registers, pre-scaled
  v2f qa[KP / 4];
#pragma unroll
  for (int c = 0; c < KP / 4; ++c) {
    const float* q = Qb + base + (size_t)(t0 + m) * KP + c * 4 + kb;
    qa[c].x = q[0] * scale;
    qa[c].y = q[1] * scale;
  }

  v8f o[5];
#pragma unroll
  for (int i = 0; i < 5; ++i) o[i] = v8f{};
  float rm[8], rl[8];
#pragma unroll
  for (int j = 0; j < 8; ++j) { rm[j] = -1e30f; rl[j] = 0.f; }

  // stage key/value tile kt into LDS buffer buf (16B per thread-iteration)
  auto stage = [&](int kt, int buf) {
    const float* gK = Kb + base + (size_t)kt * 16 * KP;
    const float* gV = Vb + base + (size_t)kt * 16 * KP;
    for (int i = threadIdx.x; i < (16 * KP) / 4; i += 256) {
      cp16_g2l(gK + i * 4, &ldsK[buf][i * 4]);
      cp16_g2l(gV + i * 4, &ldsV[buf][i * 4]);
    }
  };

  stage(0, 0);
  for (int kt = 0; kt < TSEQ / 16; ++kt) {
    const int cur = kt & 1;
    wait_async_all();
    __syncthreads();                       // tile[cur] visible, all waves past tile[cur^1]
    if (kt + 1 < TSEQ / 16) stage(kt + 1, cur ^ 1);

    // S = Q * K^T   (contraction over KP features)
    v8f s = {};
#pragma unroll
    for (int c = 0; c < KP / 4; ++c) {
      v2f bv;
      bv.x = ldsK[cur][m * KP + c * 4 + kb];
      bv.y = ldsK[cur][m * KP + c * 4 + kb + 1];
      s = wmma4(qa[c], bv, s);
    }

    // online softmax (row elements live inside one 16-lane half)
    float pr[8];
#pragma unroll
    for (int j = 0; j < 8; ++j) {
      float xv = s[j];
      float mx = xv;
      for (int off = 1; off < 16; off <<= 1) mx = fmaxf(mx, __shfl_xor(mx, off, 32));
      float mnew = fmaxf(rm[j], mx);
      float corr = __expf(rm[j] - mnew);
      float p = __expf(xv - mnew);
      float sum = p;
      for (int off = 1; off < 16; off <<= 1) sum += __shfl_xor(sum, off, 32);
      rl[j] = rl[j] * corr + sum;
      rm[j] = mnew;
      pr[j] = p;
#pragma unroll
      for (int i = 0; i < 5; ++i) o[i][j] *= corr;
    }

    // P: C-layout -> A-layout via per-wave LDS (in-wave, ds wait only)
    float* P = &ldsP[wave][0];
#pragma unroll
    for (int j = 0; j < 8; ++j) P[(mh + j) * 17 + m] = pr[j];
    __builtin_amdgcn_wave_barrier();
    asm volatile("s_wait_dscnt 0" ::: "memory");

    // O += P * V  (contraction over 16 keys, 5 output chunks of 16)
#pragma unroll
    for (int nt = 0; nt < 5; ++nt) {
      v8f acc = o[nt];
#pragma unroll
      for (int kk = 0; kk < 16; kk += 4) {
        v2f a, bv;
        a.x = P[m * 17 + kk + kb];
        a.y = P[m * 17 + kk + kb + 1];
        bv.x = ldsV[cur][(kk + kb) * KP + nt * 16 + m];
        bv.y = ldsV[cur][(kk + kb + 1) * KP + nt * 16 + m];
        acc = wmma4(a, bv, acc);
      }
      o[nt] = acc;
    }
  }

  // normalize + store
#pragma unroll
  for (int nt = 0; nt < 5; ++nt) {
    float* O = Ob + base + (size_t)t0 * KP + nt * 16;
#pragma unroll
    for (int j = 0; j < 8; ++j) O[(mh + j) * KP + m] = o[nt][j] / rl[j];
  }
}

// ---------------- head-unify: att @ Wu + bu + residual ----------------
// grid (T/16, B), block 32; wave accumulates all 5 n-tiles across 8 heads.
__global__ __launch_bounds__(32) void unify_kernel(const float* __restrict__ AO,
                                                   const float* __restrict__ WUp,
                                                   const float* __restrict__ bu,
                                                   const float* __restrict__ hres,
                                                   float* __restrict__ out) {
  const int tt = blockIdx.x * 16;
  const int b  = blockIdx.y;
  const int lane = threadIdx.x & 31;
  const int m  = lane & 15;
  const int kb = (lane >> 4) << 1;
  const int mh = (lane >> 4) << 3;
  v8f acc[5];
#pragma unroll
  for (int nt = 0; nt < 5; ++nt) acc[nt] = v8f{};
  for (int h = 0; h < HEADS; ++h) {
    const float* A  = AO + (((size_t)(b * HEADS + h)) * TSEQ + tt) * KP;
    const float* Bp = WUp + (size_t)h * KP * KP;
    for (int c = 0; c < 20; ++c) {
      v2f a;
      a.x = A[m * KP + c * 4 + kb];
      a.y = A[m * KP + c * 4 + kb + 1];
#pragma unroll
      for (int nt = 0; nt < 5; ++nt) {
        v2f bv = *(const v2f*)(Bp + ((size_t)(nt * 20 + c) * 32 + lane) * 2);
        acc[nt] = wmma4(a, bv, acc[nt]);
      }
    }
  }
#pragma unroll
  for (int nt = 0; nt < 5; ++nt) {
    const int gn = nt * 16 + m;
    const float bias = (gn < KDIM) ? bu[gn] : 0.f;
#pragma unroll
    for (int j = 0; j < 8; ++j) {
      const size_t row = (size_t)b * TSEQ + tt + mh + j;
      out[row * KP + gn] = acc[nt][j] + bias + hres[row * KP + gn];
    }
  }
}

// ---------------- layernorm over 76 dims, wave per token ----------------
__global__ __launch_bounds__(256) void ln_kernel(const float* __restrict__ in,
                                                 const float* __restrict__ s,
                                                 const float* __restrict__ bb,
                                                 float* __restrict__ out) {
  const int lane = threadIdx.x & 31, wave = threadIdx.x >> 5;
  const size_t token = (size_t)blockIdx.x * 8 + wave;
  const float* row = in + token * KP;
  const int i2 = lane + 64;
  float x0 = row[lane];
  float x1 = row[lane + 32];
  float x2 = (i2 < KDIM) ? row[i2] : 0.f;
  float sum = x0 + x1 + x2;
  for (int off = 1; off < 32; off <<= 1) sum += __shfl_xor(sum, off, 32);
  const float mean = sum * (1.f / KDIM);
  float d0 = x0 - mean, d1 = x1 - mean, d2 = (i2 < KDIM) ? x2 - mean : 0.f;
  float vs = d0 * d0 + d1 * d1 + d2 * d2;
  for (int off = 1; off < 32; off <<= 1) vs += __shfl_xor(vs, off, 32);
  const float rstd = rsqrtf(vs * (1.f / KDIM) + 1e-5f);
  float* orow = out + token * KP;
  orow[lane]      = d0 * rstd * s[lane] + bb[lane];
  orow[lane + 32] = d1 * rstd * s[lane + 32] + bb[lane + 32];
  if (i2 < KDIM)      orow[i2] = d2 * rstd * s[i2] + bb[i2];
  else if (i2 < KP)   orow[i2] = 0.f;
}

// ---------------- FFN1: relu(h1 @ W1 + b1), 5 n-tiles per wave -------------
// grid (T/16, 4, B), block 32
__global__ __launch_bounds__(32) void ffn1_kernel(const float* __restrict__ h1,
                                                  const float* __restrict__ W1p,
                                                  const float* __restrict__ b1,
                                                  float* __restrict__ ff) {
  const int tt = blockIdx.x * 16;
  const int g  = blockIdx.y;           // n-tile group: nt = g*5 .. g*5+4
  const int b  = blockIdx.z;
  const int lane = threadIdx.x & 31;
  const int m  = lane & 15;
  const int kb = (lane >> 4) << 1;
  const int mh = (lane >> 4) << 3;
  const float* A = h1 + ((size_t)b * TSEQ + tt) * KP;
  v8f acc[5];
#pragma unroll
  for (int i = 0; i < 5; ++i) acc[i] = v8f{};
  for (int c = 0; c < 20; ++c) {
    v2f a;
    a.x = A[m * KP + c * 4 + kb];
    a.y = A[m * KP + c * 4 + kb + 1];
#pragma unroll
    for (int i = 0; i < 5; ++i) {
      const int nt = g * 5 + i;
      v2f bv = *(const v2f*)(W1p + ((size_t)(nt * 20 + c) * 32 + lane) * 2);
      acc[i] = wmma4(a, bv, acc[i]);
    }
  }
#pragma unroll
  for (int i = 0; i < 5; ++i) {
    const int gn = (g * 5 + i) * 16 + m;
    const float bias = (gn < F4) ? b1[gn] : 0.f;
#pragma unroll
    for (int j = 0; j < 8; ++j) {
      const size_t row = (size_t)b * TSEQ + tt + mh + j;
      ff[row * F4P + gn] = fmaxf(acc[i][j] + bias, 0.f);
    }
  }
}

// ---------------- FFN2: ff @ W2 + b2 + residual h1, 5 n-tiles per wave -----
// grid (T/16, B), block 32
__global__ __launch_bounds__(32) void ffn2_kernel(const float* __restrict__ ff,
                                                  const float* __restrict__ W2p,
                                                  const float* __restrict__ b2,
                                                  const float* __restrict__ h1res,
                                                  float* __restrict__ out) {
  const int tt = blockIdx.x * 16;
  const int b  = blockIdx.y;
  const int lane = threadIdx.x & 31;
  const int m  = lane & 15;
  const int kb = (lane >> 4) << 1;
  const int mh = (lane >> 4) << 3;
  const float* A = ff + ((size_t)b * TSEQ + tt) * F4P;
  v8f acc[5];
#pragma unroll
  for (int nt = 0; nt < 5; ++nt) acc[nt] = v8f{};
  for (int c = 0; c < 80; ++c) {
    v2f a;
    a.x = A[m * F4P + c * 4 + kb];
    a.y = A[m * F4P + c * 4 + kb + 1];
#pragma unroll
    for (int nt = 0; nt < 5; ++nt) {
      v2f bv = *(const v2f*)(W2p + ((size_t)(nt * 80 + c) * 32 + lane) * 2);
      acc[nt] = wmma4(a, bv, acc[nt]);
    }
  }
#pragma unroll
  for (int nt = 0; nt < 5; ++nt) {
    const int gn = nt * 16 + m;
    const float bias = (gn < KDIM) ? b2[gn] : 0.f;
#pragma unroll
    for (int j = 0; j < 8; ++j) {
      const size_t row = (size_t)b * TSEQ + tt + mh + j;
      out[row * KP + gn] = acc[nt][j] + bias + h1res[row * KP + gn];
    }
  }
}

// ---------------- classifier head ----------------
__global__ __launch_bounds__(256) void head_kernel(const float* __restrict__ h,
                                                   const float* __restrict__ Wo,
                                                   const float* __restrict__ bo,
                                                   float* __restrict__ out) {
  const int i = blockIdx.x * blockDim.x + threadIdx.x;  // token*NCLASS + c
  if (i >= BATCH * TSEQ * NCLASS) return;
  const int c = i % NCLASS;
  const size_t token = i / NCLASS;
  float acc = bo[c];
  const float* row = h + token * KP;
  for (int k = 0; k < KDIM; ++k) acc += row[k] * Wo[k * NCLASS + c];
  out[i] = acc;
}

// ---------------- workspace layout (floats) ----------------
static constexpr size_t HBUF_OFF = 0;
static constexpr size_t H1_OFF   = HBUF_OFF + (size_t)BATCH * TSEQ * KP;
static constexpr size_t TMP_OFF  = H1_OFF   + (size_t)BATCH * TSEQ * KP;
static constexpr size_t Q_OFF    = TMP_OFF  + (size_t)BATCH * TSEQ * KP;
static constexpr size_t KB_OFF   = Q_OFF    + (size_t)BATCH * HEADS * TSEQ * KP;
static constexpr size_t VB_OFF   = KB_OFF   + (size_t)BATCH * HEADS * TSEQ * KP;
static constexpr size_t AO_OFF   = VB_OFF   + (size_t)BATCH * HEADS * TSEQ * KP;
static constexpr size_t FF_OFF   = AO_OFF   + (size_t)BATCH * HEADS * TSEQ * KP;
static constexpr size_t WQP_OFF  = FF_OFF   + (size_t)BATCH * TSEQ * F4P;
static constexpr size_t WKP_OFF  = WQP_OFF  + (size_t)DEPTH * HEADS * KP * KP;
static constexpr size_t WVP_OFF  = WKP_OFF  + (size_t)DEPTH * HEADS * KP * KP;
static constexpr size_t WUP_OFF  = WVP_OFF  + (size_t)DEPTH * HEADS * KP * KP;
static constexpr size_t W1P_OFF  = WUP_OFF  + (size_t)DEPTH * HEADS * KP * KP;
static constexpr size_t W2P_OFF  = W1P_OFF  + (size_t)DEPTH * KP * F4P;

extern "C" void kernel_launch(void* const* d_in, const int* in_sizes, int n_in,
                              void* d_out, int out_size, void* d_ws, size_t ws_size,
                              hipStream_t stream) {
  (void)in_sizes; (void)n_in; (void)out_size; (void)ws_size;
  const float* v    = (const float*)d_in[0];
  const float* x    = (const float*)d_in[1];
  const float* y    = (const float*)d_in[2];
  const float* z    = (const float*)d_in[3];
  const float* Whp  = (const float*)d_in[4];
  const float* bhp  = (const float*)d_in[5];
  const float* Wlm  = (const float*)d_in[6];
  const float* blm  = (const float*)d_in[7];
  const float* Wel  = (const float*)d_in[8];
  const float* bel  = (const float*)d_in[9];
  const float* Wau  = (const float*)d_in[10];
  const float* bau  = (const float*)d_in[11];
  const float* pos  = (const float*)d_in[12];
  const float* Wq   = (const float*)d_in[13];
  const float* Wk   = (const float*)d_in[14];
  const float* Wv   = (const float*)d_in[15];
  const float* Wu   = (const float*)d_in[16];
  const float* bu   = (const float*)d_in[17];
  const float* ln1s = (const float*)d_in[18];
  const float* ln1b = (const float*)d_in[19];
  const float* W1   = (const float*)d_in[20];
  const float* b1   = (const float*)d_in[21];
  const float* W2   = (const float*)d_in[22];
  const float* b2   = (const float*)d_in[23];
  const float* ln2s = (const float*)d_in[24];
  const float* ln2b = (const float*)d_in[25];
  const float* Wo   = (const float*)d_in[26];
  const float* bo   = (const float*)d_in[27];

  float* ws = (float*)d_ws;
  float* HBUF = ws + HBUF_OFF;
  float* H1   = ws + H1_OFF;
  float* TMP  = ws + TMP_OFF;
  float* QB   = ws + Q_OFF;
  float* KB   = ws + KB_OFF;
  float* VB   = ws + VB_OFF;
  float* AOB  = ws + AO_OFF;
  float* FFB  = ws + FF_OFF;
  float* WQP  = ws + WQP_OFF;
  float* WKP  = ws + WKP_OFF;
  float* WVP  = ws + WVP_OFF;
  float* WUP  = ws + WUP_OFF;
  float* W1P  = ws + W1P_OFF;
  float* W2P  = ws + W2P_OFF;

  // weight packing into WMMA B-chunk order
  {
    const int tq = DEPTH * HEADS * KP * KP;
    pack_qkv_w<<<(tq + 255) / 256, 256, 0, stream>>>(Wq, WQP);
    pack_qkv_w<<<(tq + 255) / 256, 256, 0, stream>>>(Wk, WKP);
    pack_qkv_w<<<(tq + 255) / 256, 256, 0, stream>>>(Wv, WVP);
    pack_wu<<<(tq + 255) / 256, 256, 0, stream>>>(Wu, WUP);
    const int t1 = DEPTH * KP * F4P;
    pack_w1<<<(t1 + 255) / 256, 256, 0, stream>>>(W1, W1P);
    pack_w2<<<(t1 + 255) / 256, 256, 0, stream>>>(W2, W2P);
  }

  // embed + pos
  embed_kernel<<<BATCH * TSEQ, 128, 0, stream>>>(v, x, y, z, Whp, bhp, Wlm, blm,
                                                 Wel, bel, Wau, bau, pos, HBUF);

  const dim3 gProj(TSEQ / 16, HEADS, BATCH);
  const dim3 gAttn(TSEQ / 128, HEADS, BATCH);
  const dim3 gUni(TSEQ / 16, BATCH);
  const dim3 gF1(TSEQ / 16, 4, BATCH);
  const dim3 gF2(TSEQ / 16, BATCH);

  for (int d = 0; d < DEPTH; ++d) {
    const float* wqp = WQP + (size_t)d * HEADS * KP * KP;
    const float* wkp = WKP + (size_t)d * HEADS * KP * KP;
    const float* wvp = WVP + (size_t)d * HEADS * KP * KP;
    const float* wup = WUP + (size_t)d * HEADS * KP * KP;
    const float* w1p = W1P + (size_t)d * KP * F4P;
    const float* w2p = W2P + (size_t)d * F4P * KP;

    qkv_kernel<<<gProj, 32, 0, stream>>>(HBUF, wqp, QB);
    qkv_kernel<<<gProj, 32, 0, stream>>>(HBUF, wkp, KB);
    qkv_kernel<<<gProj, 32, 0, stream>>>(HBUF, wvp, VB);

    attn_kernel<<<gAttn, 256, 0, stream>>>(QB, KB, VB, AOB);

    unify_kernel<<<gUni, 32, 0, stream>>>(AOB, wup, bu + d * KDIM, HBUF, TMP);
    ln_kernel<<<BATCH * TSEQ / 8, 256, 0, stream>>>(TMP, ln1s + d * KDIM,
                                                    ln1b + d * KDIM, H1);

    ffn1_kernel<<<gF1, 32, 0, stream>>>(H1, w1p, b1 + d * F4, FFB);
    ffn2_kernel<<<gF2, 32, 0, stream>>>(FFB, w2p, b2 + d * KDIM, H1, TMP);
    ln_kernel<<<BATCH * TSEQ / 8, 256, 0, stream>>>(TMP, ln2s + d * KDIM,
                                                    ln2b + d * KDIM, HBUF);
  }

  head_kernel<<<(BATCH * TSEQ * NCLASS + 255) / 256, 256, 0, stream>>>(
      HBUF, Wo, bo, (float*)d_out);
}